// HistogramSelfAttention_81011673137701
// MI455X (gfx1250) — compile-verified
//
#include <hip/hip_runtime.h>
#include <hip/hip_bf16.h>
#include <math.h>

// ---------------- problem constants (from reference setup_inputs) ----------
#define Bb    2
#define Cc    64
#define HALF  32
#define Hh    256
#define Ww    256
#define Nn    65536          // Hh*Ww
#define HEADS 8
#define CO    320            // 5*Cc
#define Gg    8192           // Nn/HEADS
#define KSL   8              // K-slices for the score GEMM

typedef __attribute__((ext_vector_type(2))) float v2f;
typedef __attribute__((ext_vector_type(8))) float v8f;
typedef __attribute__((ext_vector_type(2))) int   v2i;

#if defined(__has_builtin)
#if __has_builtin(__builtin_amdgcn_wmma_f32_16x16x4_f32)
#define HAVE_WMMA_F32X4 1
#endif
#if __has_builtin(__builtin_amdgcn_global_load_async_to_lds_b64)
#define HAVE_ASYNC_LD 1
#endif
#if __has_builtin(__builtin_amdgcn_global_store_async_from_lds_b64)
#define HAVE_ASYNC_ST 1
#endif
#endif

// builtin param types (from compiler diagnostic): b64 <-> V2i pointers,
// global = addrspace(1), LDS = addrspace(3)
typedef __attribute__((address_space(1))) v2i* g2ip;
typedef __attribute__((address_space(3))) v2i* l2ip;

__device__ __forceinline__ void wait_async0() {
#if defined(__has_builtin) && __has_builtin(__builtin_amdgcn_s_wait_asynccnt)
  __builtin_amdgcn_s_wait_asynccnt(0);
#else
  asm volatile("s_wait_asynccnt 0x0" ::: "memory");
#endif
}

// async copy 8B global -> LDS (per-lane addresses), falls back to VGPR path
__device__ __forceinline__ void cp_g2l_b64(unsigned long long* lds,
                                           const unsigned long long* g) {
#ifdef HAVE_ASYNC_LD
  __builtin_amdgcn_global_load_async_to_lds_b64(
      (g2ip)(unsigned long long)g, (l2ip)(unsigned)(unsigned long long)lds, 0, 0);
#else
  *lds = *g;
#endif
}
__device__ __forceinline__ void cp_l2g_b64(unsigned long long* g,
                                           const unsigned long long* lds) {
#ifdef HAVE_ASYNC_ST
  __builtin_amdgcn_global_store_async_from_lds_b64(
      (g2ip)(unsigned long long)g, (l2ip)(unsigned)(unsigned long long)lds, 0, 0);
#else
  *g = *lds;
#endif
}

// D(16x16,f32) = A(16x4,f32) * B(4x16,f32) + C   -- full fp32 precision path
__device__ __forceinline__ v8f wmma4(v2f a, v2f b, v8f c) {
#ifdef HAVE_WMMA_F32X4
  return __builtin_amdgcn_wmma_f32_16x16x4_f32(false, a, false, b, (short)0, c,
                                               false, false);
#else
  return c;
#endif
}

// monotone float <-> uint mapping for bitonic ordering
__device__ __forceinline__ unsigned f2sort(float f) {
  unsigned u = __float_as_uint(f);
  return (u & 0x80000000u) ? ~u : (u | 0x80000000u);
}
__device__ __forceinline__ float sort2f(unsigned s) {
  unsigned u = (s & 0x80000000u) ? (s & 0x7fffffffu) : ~s;
  return __uint_as_float(u);
}

// ===========================================================================
// Stage A: sort lower-half channels along h, then along w (256-elt bitonic)
// ===========================================================================
__global__ __launch_bounds__(256) void sort_h_kernel(const float* __restrict__ x,
                                                     float* __restrict__ xs,
                                                     int* __restrict__ ih) {
  __shared__ unsigned long long sm[256];
  int col = blockIdx.x;                 // (b*HALF + c)*Ww + w
  int w   = col & (Ww - 1);
  int bc  = col >> 8;                   // b*HALF + c
  int b   = bc / HALF, c = bc % HALF;
  const float* src = x + ((size_t)(b * Cc + c)) * Nn + w;  // stride Ww over h
  int t = threadIdx.x;
  float f = src[(size_t)t * Ww];
  sm[t] = ((unsigned long long)f2sort(f) << 32) | (unsigned)t;
  __syncthreads();
  for (int k = 2; k <= 256; k <<= 1)
    for (int j = k >> 1; j > 0; j >>= 1) {
      int ixj = t ^ j;
      if (ixj > t) {
        unsigned long long a = sm[t], bv = sm[ixj];
        bool up = ((t & k) == 0);
        if ((a > bv) == up) { sm[t] = bv; sm[ixj] = a; }
      }
      __syncthreads();
    }
  unsigned long long e = sm[t];
  xs[((size_t)(b * Cc + c)) * Nn + (size_t)t * Ww + w] = sort2f((unsigned)(e >> 32));
  ih[((size_t)bc) * Nn + (size_t)t * Ww + w] = (int)(e & 0xffffffffu);
}

__global__ __launch_bounds__(256) void sort_w_kernel(float* __restrict__ xs,
                                                     int* __restrict__ iw) {
  __shared__ unsigned long long sm[256];
  int row = blockIdx.x;                 // (b*HALF + c)*Hh + h
  int h   = row & (Hh - 1);
  int bc  = row >> 8;
  int b   = bc / HALF, c = bc % HALF;
  float* src = xs + ((size_t)(b * Cc + c)) * Nn + (size_t)h * Ww;
  int t = threadIdx.x;
  sm[t] = ((unsigned long long)f2sort(src[t]) << 32) | (unsigned)t;
  __syncthreads();
  for (int k = 2; k <= 256; k <<= 1)
    for (int j = k >> 1; j > 0; j >>= 1) {
      int ixj = t ^ j;
      if (ixj > t) {
        unsigned long long a = sm[t], bv = sm[ixj];
        bool up = ((t & k) == 0);
        if ((a > bv) == up) { sm[t] = bv; sm[ixj] = a; }
      }
      __syncthreads();
    }
  unsigned long long e = sm[t];
  src[t] = sort2f((unsigned)(e >> 32));
  iw[((size_t)bc) * Nn + (size_t)h * Ww + t] = (int)(e & 0xffffffffu);
}

__global__ __launch_bounds__(256) void copy_upper_kernel(const float* __restrict__ x,
                                                         float* __restrict__ xs) {
  size_t id = (size_t)blockIdx.x * 256 + threadIdx.x;   // over B*HALF*Nn
  if (id >= (size_t)Bb * HALF * Nn) return;
  size_t p = id & (Nn - 1);
  size_t bc = id >> 16;
  int b = (int)(bc / HALF), c = (int)(bc % HALF);
  size_t o = ((size_t)(b * Cc + HALF + c)) * Nn + p;
  __builtin_prefetch(&x[o + 4096], 0, 1);
  xs[o] = x[o];
}

// ===========================================================================
// fp32 WMMA GEMM:  Y[b, o, n] = sum_c Wm[o,c] * X[b, c, n]   (K = Cc = 64)
// ===========================================================================
__global__ __launch_bounds__(256) void gemm_wmma_kernel(const float* __restrict__ Wm,
                                                        const float* __restrict__ X,
                                                        float* __restrict__ Y, int M) {
  int lane = threadIdx.x & 31;
  int wave = (int)(((size_t)blockIdx.x * 256 + threadIdx.x) >> 5);
  int ntn = Nn / 16, mt = M / 16;
  int nt = wave % ntn; int r = wave / ntn;
  int ot = r % mt;     int b = r / mt;
  if (b >= Bb) return;
  const float* Wp = Wm + (size_t)(ot * 16 + (lane & 15)) * Cc;
  const float* Xp = X + (size_t)b * Cc * Nn + (size_t)nt * 16 + (lane & 15);
  int khi = (lane >> 4) << 1;   // lanes 0-15 -> K{0,1}; 16-31 -> K{2,3}
  v8f acc = {};
  #pragma unroll
  for (int k0 = 0; k0 < Cc; k0 += 4) {
    v2f a, bv;
    a.x = Wp[k0 + khi];
    a.y = Wp[k0 + khi + 1];
    bv.x = Xp[(size_t)(k0 + khi) * Nn];
    bv.y = Xp[(size_t)(k0 + khi + 1) * Nn];
    acc = wmma4(a, bv, acc);
  }
  float* Yp = Y + ((size_t)b * M + ot * 16) * Nn + (size_t)nt * 16 + (lane & 15);
  int mbase = (lane >> 4) << 3;
  #pragma unroll
  for (int rr = 0; rr < 8; rr++) Yp[(size_t)(mbase + rr) * Nn] = acc[rr];
}

// ===========================================================================
// depthwise 3x3, pad 1  (memory-bound stream; qkv nearly fits the 192MB L2)
// ===========================================================================
__global__ __launch_bounds__(256) void dwconv_kernel(const float* __restrict__ in,
                                                     const float* __restrict__ wdw,
                                                     float* __restrict__ out) {
  size_t id = (size_t)blockIdx.x * 256 + threadIdx.x;   // over B*CO*Nn
  if (id >= (size_t)Bb * CO * Nn) return;
  int w = (int)(id & (Ww - 1));
  int h = (int)((id >> 8) & (Hh - 1));
  size_t bc = id >> 16;
  int ch = (int)(bc % CO);
  const float* wp = wdw + ch * 9;
  const float* ip = in + (bc << 16);
  float s = 0.f;
  #pragma unroll
  for (int dy = -1; dy <= 1; dy++) {
    int hh = h + dy;
    if (hh < 0 || hh >= Hh) continue;
    #pragma unroll
    for (int dx = -1; dx <= 1; dx++) {
      int ww2 = w + dx;
      if (ww2 < 0 || ww2 >= Ww) continue;
      s += wp[(dy + 1) * 3 + (dx + 1)] * ip[(size_t)hh * Ww + ww2];
    }
  }
  out[id] = s;
}

// ===========================================================================
// Stage D: 65536-length argsort of v per (b,c): hybrid bitonic on u64 keys
//   LDS chunks are filled/drained with gfx1250 async global<->LDS ops.
// ===========================================================================
__global__ __launch_bounds__(256) void pack_keys_kernel(const float* __restrict__ qkvb,
                                                        unsigned long long* __restrict__ keys) {
  size_t id = (size_t)blockIdx.x * 256 + threadIdx.x;   // over B*Cc*Nn
  if (id >= (size_t)Bb * Cc * Nn) return;
  size_t p = id & (Nn - 1);
  size_t bc = id >> 16;                                  // b*Cc + c
  int b = (int)(bc >> 6), c = (int)(bc & 63);
  float v = qkvb[((size_t)b * CO + 4 * Cc + c) * Nn + p];
  keys[id] = ((unsigned long long)f2sort(v) << 32) | (unsigned)p;
}

__global__ __launch_bounds__(256) void bitonic_local_sort(unsigned long long* __restrict__ keys) {
  __shared__ unsigned long long sm[4096];
  size_t base = (size_t)blockIdx.x * 4096;
  for (int e = threadIdx.x; e < 4096; e += 256) cp_g2l_b64(&sm[e], &keys[base + e]);
  wait_async0();
  __syncthreads();
  for (int k = 2; k <= 4096; k <<= 1)
    for (int j = k >> 1; j > 0; j >>= 1) {
      for (int e = threadIdx.x; e < 2048; e += 256) {
        int i = ((e / j) * (j << 1)) + (e % j);
        int p = i + j;
        unsigned long long a = sm[i], bv = sm[p];
        bool up = ((i & k) == 0);
        if ((a > bv) == up) { sm[i] = bv; sm[p] = a; }
      }
      __syncthreads();
    }
  for (int e = threadIdx.x; e < 4096; e += 256) cp_l2g_b64(&keys[base + e], &sm[e]);
}

__global__ __launch_bounds__(256) void bitonic_global_pass(unsigned long long* __restrict__ keys,
                                                           int k, int j) {
  size_t id = (size_t)blockIdx.x * 256 + threadIdx.x;   // over B*Cc*(Nn/2)
  if (id >= (size_t)Bb * Cc * (Nn / 2)) return;
  size_t row = id / (Nn / 2);
  int e = (int)(id % (Nn / 2));
  int i = ((e / j) * (j << 1)) + (e % j);
  int p = i + j;
  unsigned long long* r = keys + row * Nn;
  unsigned long long a = r[i], bv = r[p];
  bool up = ((i & k) == 0);
  if ((a > bv) == up) { r[i] = bv; r[p] = a; }
}

__global__ __launch_bounds__(256) void bitonic_local_merge(unsigned long long* __restrict__ keys,
                                                           int k) {
  __shared__ unsigned long long sm[4096];
  size_t base = (size_t)blockIdx.x * 4096;
  int row_off = (int)(base & (Nn - 1));        // chunk start within row
  bool up = ((row_off & k) == 0);              // k >= 8192 > chunk: constant dir
  for (int e = threadIdx.x; e < 4096; e += 256) cp_g2l_b64(&sm[e], &keys[base + e]);
  wait_async0();
  __syncthreads();
  for (int j = 2048; j > 0; j >>= 1) {
    for (int e = threadIdx.x; e < 2048; e += 256) {
      int i = ((e / j) * (j << 1)) + (e % j);
      int p = i + j;
      unsigned long long a = sm[i], bv = sm[p];
      if ((a > bv) == up) { sm[i] = bv; sm[p] = a; }
    }
    __syncthreads();
  }
  for (int e = threadIdx.x; e < 4096; e += 256) cp_l2g_b64(&keys[base + e], &sm[e]);
}

// gather q1/k1/q2/k2/v by idx; build inverse permutation
__global__ __launch_bounds__(256) void gather_kernel(const unsigned long long* __restrict__ keys,
                                                     const float* __restrict__ qkvb,
                                                     float* __restrict__ gath,
                                                     int* __restrict__ invidx) {
  size_t id = (size_t)blockIdx.x * 256 + threadIdx.x;   // over B*Cc*Nn
  if (id >= (size_t)Bb * Cc * Nn) return;
  size_t p = id & (Nn - 1);
  size_t bc = id >> 16;
  int b = (int)(bc >> 6), c = (int)(bc & 63);
  unsigned m = (unsigned)(keys[id] & 0xffffffffu);
  invidx[id - p + m] = (int)p;                          // inv[bc][m] = p
  size_t sb = (size_t)b * CO * Nn;
  #pragma unroll
  for (int rgn = 0; rgn < 5; rgn++) {
    size_t ch = (size_t)(rgn * Cc + c) * Nn;
    gath[sb + ch + p] = qkvb[sb + ch + m];
  }
}

// ===========================================================================
// Attention. split(first=True):  row i=(cp*8+h2), elem g -> ch 8h1+cp, n=h2*Gg+g
//            split(first=False): elem g -> ch 8h1+cp, n = h2 + 8*g
// ===========================================================================
__global__ __launch_bounds__(256) void sumsq_kernel(const float* __restrict__ gath,
                                                    float* __restrict__ sums) {
  __shared__ float red[256];
  int row = blockIdx.x;                  // arr(4) * [b][h1][i]
  int arr = row >> 10;
  int r2 = row & 1023;
  int i = r2 & 63, bh = r2 >> 6;
  int h1 = bh & 7, b = bh >> 3;
  int cp = i >> 3, h2 = i & 7;
  bool first = (arr < 2);
  const float* src = gath + ((size_t)b * CO + (size_t)arr * Cc + 8 * h1 + cp) * Nn;
  float s = 0.f;
  for (int g = threadIdx.x; g < Gg; g += 256) {
    float v = first ? src[(size_t)h2 * Gg + g] : src[h2 + 8 * (size_t)g];
    s += v * v;
  }
  red[threadIdx.x] = s;
  __syncthreads();
  for (int o = 128; o > 0; o >>= 1) {
    if ((int)threadIdx.x < o) red[threadIdx.x] += red[threadIdx.x + o];
    __syncthreads();
  }
  if (threadIdx.x == 0) sums[row] = red[0];
}

// partial scores over one K-slice of 1024 (deterministic split, no atomics)
__global__ __launch_bounds__(256) void scores_kernel(const float* __restrict__ gath,
                                                     float* __restrict__ Spart) {
  int lane = threadIdx.x & 31, wv = threadIdx.x >> 5;
  int blk = blockIdx.x >> 3;             // (attn*2 + b)*8 + h1
  int ks  = blockIdx.x & (KSL - 1);      // K slice
  int h1 = blk & 7, r = blk >> 3;
  int b = r & 1, attn = r >> 1;
  int qreg = attn ? 2 : 0;
  const float* Qb = gath + ((size_t)b * CO + (size_t)qreg * Cc + 8 * h1) * Nn;
  const float* Kb = gath + ((size_t)b * CO + (size_t)(qreg + 1) * Cc + 8 * h1) * Nn;
  float* Sp = Spart + ((size_t)blk * KSL + ks) * 4096;
  int khi = (lane >> 4) << 1;
  int gbeg = ks * (Gg / KSL), gend = gbeg + (Gg / KSL);
  for (int tile = wv; tile < 16; tile += 8) {           // 4x4 tiles of 16x16
    int it = tile >> 2, jt = tile & 3;
    int mi = it * 16 + (lane & 15);  int cpi = mi >> 3, h2i = mi & 7;
    int mj = jt * 16 + (lane & 15);  int cpj = mj >> 3, h2j = mj & 7;
    const float* qp = Qb + (size_t)cpi * Nn;
    const float* kp = Kb + (size_t)cpj * Nn;
    v8f acc = {};
    if (attn == 0) {
      const float* q0 = qp + (size_t)h2i * Gg;
      const float* k0p = kp + (size_t)h2j * Gg;
      for (int g0 = gbeg + khi; g0 < gend; g0 += 4) {
        __builtin_prefetch(&q0[g0 + 256], 0, 1);
        v2f a, bv;
        a.x = q0[g0];  a.y = q0[g0 + 1];
        bv.x = k0p[g0]; bv.y = k0p[g0 + 1];
        acc = wmma4(a, bv, acc);
      }
    } else {
      for (int g0 = gbeg + khi; g0 < gend; g0 += 4) {
        v2f a, bv;
        a.x = qp[h2i + 8 * (size_t)g0];  a.y = qp[h2i + 8 * (size_t)(g0 + 1)];
        bv.x = kp[h2j + 8 * (size_t)g0]; bv.y = kp[h2j + 8 * (size_t)(g0 + 1)];
        acc = wmma4(a, bv, acc);
      }
    }
    int jcol = jt * 16 + (lane & 15);
    int mbase = (lane >> 4) << 3;
    #pragma unroll
    for (int rr = 0; rr < 8; rr++)
      Sp[(size_t)(it * 16 + mbase + rr) * 64 + jcol] = acc[rr];
  }
}

__global__ __launch_bounds__(64) void softmax_kernel(const float* __restrict__ Spart,
                                                     const float* __restrict__ sums,
                                                     const float* __restrict__ temp,
                                                     float* __restrict__ P) {
  int blk = blockIdx.x;                  // (attn*2 + b)*8 + h1
  int h1 = blk & 7, r = blk >> 3;
  int b = r & 1, attn = r >> 1;
  int qarr = attn ? 2 : 0;
  const float* nq = sums + ((size_t)(qarr * Bb + b) * HEADS + h1) * 64;
  const float* nk = sums + ((size_t)((qarr + 1) * Bb + b) * HEADS + h1) * 64;
  int i = threadIdx.x;
  float qn = fmaxf(sqrtf(nq[i]), 1e-12f);
  float t = temp[h1];
  const float* Sp = Spart + (size_t)blk * KSL * 4096 + (size_t)i * 64;
  float* Pp = P + (size_t)blk * 4096 + (size_t)i * 64;
  float row[64];
  float mx = -INFINITY;
  for (int j = 0; j < 64; j++) {
    float acc = 0.f;
    #pragma unroll
    for (int s = 0; s < KSL; s++) acc += Sp[(size_t)s * 4096 + j]; // fixed order
    float kn = fmaxf(sqrtf(nk[j]), 1e-12f);
    float v = (acc / (qn * kn)) * t;
    row[j] = v;
    mx = fmaxf(mx, v);
  }
  float sum = 0.f;
  #pragma unroll 8
  for (int j = 0; j < 64; j++) { row[j] = expf(row[j] - mx); sum += row[j]; }
  float inv = 1.f / sum;
  #pragma unroll 8
  for (int j = 0; j < 64; j++) Pp[j] = row[j] * inv;
}

__global__ __launch_bounds__(256) void pv_kernel(const float* __restrict__ P,
                                                 const float* __restrict__ gath,
                                                 float* __restrict__ aout) {
  int lane = threadIdx.x & 31;
  int gwave = (int)(((size_t)blockIdx.x * 256 + threadIdx.x) >> 5);  // 65536 waves
  int gt = gwave & 511; int r = gwave >> 9;
  int it = r & 3; r >>= 2;
  int h1 = r & 7; r >>= 3;
  int b = r & 1; int attn = r >> 1;
  const float* Pp = P + (size_t)((attn * 2 + b) * 8 + h1) * 4096;
  const float* Vb = gath + ((size_t)b * CO + 4 * Cc + 8 * h1) * Nn;
  int mi = it * 16 + (lane & 15);       // A row (i)
  int gcol = gt * 16 + (lane & 15);     // B col (g)
  int khi = (lane >> 4) << 1;
  v8f acc = {};
  #pragma unroll
  for (int j0 = 0; j0 < 64; j0 += 4) {
    int ja = j0 + khi;
    v2f a, bv;
    a.x = Pp[(size_t)mi * 64 + ja];
    a.y = Pp[(size_t)mi * 64 + ja + 1];
    const float* v0 = Vb + (size_t)(ja >> 3) * Nn;
    const float* v1 = Vb + (size_t)((ja + 1) >> 3) * Nn;
    if (attn == 0) {
      bv.x = v0[(size_t)(ja & 7) * Gg + gcol];
      bv.y = v1[(size_t)((ja + 1) & 7) * Gg + gcol];
    } else {
      bv.x = v0[(ja & 7) + 8 * (size_t)gcol];
      bv.y = v1[((ja + 1) & 7) + 8 * (size_t)gcol];
    }
    acc = wmma4(a, bv, acc);
  }
  // output layout (both attns): ch = 8*h1 + (i>>3), n = (i&7)*Gg + g
  float* Ob = aout + (size_t)attn * ((size_t)Bb * Cc * Nn) +
              ((size_t)b * Cc + 8 * h1) * Nn;
  int mbase = (lane >> 4) << 3;
  #pragma unroll
  for (int rr = 0; rr < 8; rr++) {
    int i = it * 16 + mbase + rr;
    Ob[(size_t)(i >> 3) * Nn + (size_t)(i & 7) * Gg + gcol] = acc[rr];
  }
}

// o[b,c,m] = a1[b,c,inv[m]] * a2[b,c,inv[m]]  (un-sorts back to pixel order)
__global__ __launch_bounds__(256) void combine_kernel(const float* __restrict__ aout,
                                                      const int* __restrict__ invidx,
                                                      float* __restrict__ comb) {
  size_t id = (size_t)blockIdx.x * 256 + threadIdx.x;   // over B*Cc*Nn
  if (id >= (size_t)Bb * Cc * Nn) return;
  size_t bc = id >> 16;
  int p = invidx[id];
  size_t s = (bc << 16) + (unsigned)p;
  comb[id] = aout[s] * aout[(size_t)Bb * Cc * Nn + s];
}

// inverse spatial permutations on the lower-half channels
__global__ __launch_bounds__(256) void unsort_w_kernel(const float* __restrict__ outp,
                                                       const int* __restrict__ iw,
                                                       float* __restrict__ tmp) {
  size_t id = (size_t)blockIdx.x * 256 + threadIdx.x;   // over B*HALF*Nn
  if (id >= (size_t)Bb * HALF * Nn) return;
  int w = (int)(id & 255);
  int h = (int)((id >> 8) & 255);
  size_t bc = id >> 16;
  int b = (int)(bc / HALF), c = (int)(bc % HALF);
  float v = outp[((size_t)(b * Cc + c)) * Nn + (size_t)h * Ww + w];
  int dw = iw[id];
  tmp[(bc << 16) + (size_t)h * Ww + dw] = v;
}

__global__ __launch_bounds__(256) void unsort_h_kernel(const float* __restrict__ tmp,
                                                       const int* __restrict__ ih,
                                                       float* __restrict__ outp) {
  size_t id = (size_t)blockIdx.x * 256 + threadIdx.x;   // over B*HALF*Nn
  if (id >= (size_t)Bb * HALF * Nn) return;
  int w = (int)(id & 255);
  int p = (int)((id >> 8) & 255);
  size_t bc = id >> 16;
  int b = (int)(bc / HALF), c = (int)(bc % HALF);
  int dh = ih[id];
  outp[((size_t)(b * Cc + c)) * Nn + (size_t)dh * Ww + w] =
      tmp[(bc << 16) + (size_t)p * Ww + w];
}

// ===========================================================================
extern "C" void kernel_launch(void* const* d_in, const int* in_sizes, int n_in,
                              void* d_out, int out_size, void* d_ws, size_t ws_size,
                              hipStream_t stream) {
  (void)in_sizes; (void)n_in; (void)out_size; (void)ws_size;
  const float* x      = (const float*)d_in[0];   // (2,64,256,256)
  const float* temp   = (const float*)d_in[1];   // (8,1,1)
  const float* w_qkv  = (const float*)d_in[2];   // (320,64)
  const float* w_dw   = (const float*)d_in[3];   // (320,1,3,3)
  const float* w_proj = (const float*)d_in[4];   // (64,64)
  float* outp = (float*)d_out;

  char* ws = (char*)d_ws;
  const size_t SZ_XC  = (size_t)Bb * Cc * Nn * 4;        // 33.5 MB
  const size_t SZ_HC  = (size_t)Bb * HALF * Nn * 4;      // 16.8 MB
  const size_t SZ_QKV = (size_t)Bb * CO * Nn * 4;        // 167.8 MB
  float* xs   = (float*)(ws);
  int*   ih   = (int*)  (ws + SZ_XC);
  int*   iw   = (int*)  (ws + SZ_XC + SZ_HC);
  float* qkva = (float*)(ws + SZ_XC + 2 * SZ_HC);
  float* qkvb = (float*)(ws + SZ_XC + 2 * SZ_HC + SZ_QKV);
  unsigned long long* keys =
      (unsigned long long*)(ws + SZ_XC + 2 * SZ_HC + 2 * SZ_QKV);        // 67 MB
  int* invidx = (int*)(ws + SZ_XC + 2 * SZ_HC + 2 * SZ_QKV + 2 * SZ_XC); // 33.5 MB
  // aliases of dead regions (producer/consumer ordering keeps this safe):
  float* aout  = qkvb;                                    // 67 MB after gather
  float* sums  = (float*)((char*)qkvb + 2 * SZ_XC);       // 4*2*8*64 floats
  float* Spart = (float*)((char*)qkvb + 2 * SZ_XC + (1 << 20)); // 32*8*4096 f
  float* P     = Spart + (size_t)32 * KSL * 4096;
  float* comb  = (float*)keys;                            // keys dead after gather
  float* tmph  = (float*)((char*)keys + SZ_XC);           // lower-half temp

  // Stage A: spatial sorts on lower-half channels
  sort_h_kernel<<<Bb * HALF * Ww, 256, 0, stream>>>(x, xs, ih);
  sort_w_kernel<<<Bb * HALF * Hh, 256, 0, stream>>>(xs, iw);
  copy_upper_kernel<<<(Bb * HALF * Nn + 255) / 256, 256, 0, stream>>>(x, xs);

  // Stage B: qkv projection (WMMA f32 16x16x4), M=320
  {
    int waves = Bb * (CO / 16) * (Nn / 16);
    gemm_wmma_kernel<<<(waves * 32 + 255) / 256, 256, 0, stream>>>(w_qkv, xs, qkva, CO);
  }
  // Stage C: depthwise 3x3
  dwconv_kernel<<<((size_t)Bb * CO * Nn + 255) / 256, 256, 0, stream>>>(qkva, w_dw, qkvb);

  // Stage D: 65536-length argsorts of v (hybrid bitonic, async LDS staging)
  pack_keys_kernel<<<((size_t)Bb * Cc * Nn + 255) / 256, 256, 0, stream>>>(qkvb, keys);
  bitonic_local_sort<<<Bb * Cc * (Nn / 4096), 256, 0, stream>>>(keys);
  for (int k = 8192; k <= Nn; k <<= 1) {
    for (int j = k >> 1; j >= 4096; j >>= 1)
      bitonic_global_pass<<<((size_t)Bb * Cc * (Nn / 2) + 255) / 256, 256, 0, stream>>>(keys, k, j);
    bitonic_local_merge<<<Bb * Cc * (Nn / 4096), 256, 0, stream>>>(keys, k);
  }
  gather_kernel<<<((size_t)Bb * Cc * Nn + 255) / 256, 256, 0, stream>>>(keys, qkvb, qkva, invidx);

  // Stage E: dual histogram attention
  sumsq_kernel<<<4 * Bb * HEADS * 64, 256, 0, stream>>>(qkva, sums);
  scores_kernel<<<2 * Bb * HEADS * KSL, 256, 0, stream>>>(qkva, Spart);
  softmax_kernel<<<2 * Bb * HEADS, 64, 0, stream>>>(Spart, sums, temp, P);
  pv_kernel<<<(2 * Bb * HEADS * 4 * 512 * 32) / 256, 256, 0, stream>>>(P, qkva, aout);

  // Stage F: combine, project, and undo the spatial sorts
  combine_kernel<<<((size_t)Bb * Cc * Nn + 255) / 256, 256, 0, stream>>>(aout, invidx, comb);
  {
    int waves = Bb * (Cc / 16) * (Nn / 16);
    gemm_wmma_kernel<<<(waves * 32 + 255) / 256, 256, 0, stream>>>(w_proj, comb, outp, Cc);
  }
  unsort_w_kernel<<<(Bb * HALF * Nn + 255) / 256, 256, 0, stream>>>(outp, iw, tmph);
  unsort_h_kernel<<<(Bb * HALF * Nn + 255) / 256, 256, 0, stream>>>(tmph, ih, outp);
}